// ActorCritic_90726889161550
// MI455X (gfx1250) — compile-verified
//
#include <hip/hip_runtime.h>
#include <math.h>

// ---------------- problem constants (from reference) ----------------
#define J_DIM   128
#define M_DIM   512
#define F_DIM   256
#define A_DIM   32
#define HID_DIM 400

#define GAMMA_C  0.999f
#define LAMBDA_C 0.95f
#define TEMP_C   0.001f
#define LN_EPS_C 0.001f

// ---------------- tiling ----------------
#define TM  16              // rows per workgroup (one 16-row WMMA M-tile)
#define LDH 404             // LDS activation row stride (floats); 404%64=20 -> conflict-free
#define MLP_THREADS 256     // 8 wave32s
#define NWAVES (MLP_THREADS / 32)
#define PANEL_K 16          // K rows of W staged per panel
#define PANEL_FLOATS (PANEL_K * HID_DIM)             // 6400 floats, contiguous in W
#define PANEL_VECS   (PANEL_FLOATS / 4)              // 1600 x b128
#define VECS_PER_THR ((PANEL_VECS + MLP_THREADS - 1) / MLP_THREADS)  // 7 (wave-uniform)
#define NT_HID (HID_DIM / 16)                        // 25 column tiles
#define MAX_TILES 4                                  // ceil(25/8)

typedef float v2f __attribute__((ext_vector_type(2)));
typedef float v8f __attribute__((ext_vector_type(8)));
typedef int   v4i __attribute__((vector_size(16)));  // matches async builtin param

#define AS_GLOBAL __attribute__((address_space(1)))
#define AS_LDS    __attribute__((address_space(3)))

// ---- async global->LDS path (gfx1250), with portable fallback ----
#if defined(__HIP_DEVICE_COMPILE__) && defined(__gfx1250__) && \
    __has_builtin(__builtin_amdgcn_global_load_async_to_lds_b128)
#define HAVE_ASYNC 1
#else
#define HAVE_ASYNC 0
#endif

#if HAVE_ASYNC
#if __has_builtin(__builtin_amdgcn_s_wait_asynccnt)
#define WAIT_ASYNC(n) __builtin_amdgcn_s_wait_asynccnt(n)
#else
#define WAIT_ASYNC(n) asm volatile("s_wait_asynccnt %0" ::"n"(n))
#endif
#else
#define WAIT_ASYNC(n) do { } while (0)
#endif

struct MlpParams {
  const float* W[5];   // W1..W4 (din x dout row-major), Wf
  const float* b[5];
  const float* g[4];   // LN gain
  const float* be[4];  // LN bias
};

__device__ __forceinline__ v8f wmma_f32(v2f a, v2f b, v8f c) {
  // V_WMMA_F32_16X16X4_F32 : D(16x16,f32) = A(16x4,f32) * B(4x16,f32) + C
  return __builtin_amdgcn_wmma_f32_16x16x4_f32(false, a, false, b, (short)0, c,
                                               false, false);
}

// Cooperative copy of one contiguous W panel (16 x 400 floats) into LDS.
// Exactly VECS_PER_THR b128 async ops per thread (tail index clamped so the
// per-wave ASYNCcnt increment is a compile-time constant for s_wait_asynccnt).
__device__ __forceinline__ void copy_panel(const float* __restrict__ src,
                                           float* dst) {
  const int tid = threadIdx.x;
#pragma unroll
  for (int v = 0; v < VECS_PER_THR; ++v) {
    int i = tid + v * MLP_THREADS;
    if (i >= PANEL_VECS) i = PANEL_VECS - 1;  // duplicate tail write: benign
#if HAVE_ASYNC
    __builtin_amdgcn_global_load_async_to_lds_b128(
        (AS_GLOBAL v4i*)(src + 4 * (size_t)i),
        (AS_LDS v4i*)(dst + 4 * i), 0, 0);
#else
    *(float4*)(dst + 4 * i) = *(const float4*)(src + 4 * i);
#endif
  }
}

// One K-panel of the GEMM — completely branch-free. Tile indices are CLAMPED
// (s_min) instead of guarded: waves whose 4th slot exceeds the 25 tiles just
// recompute tile 24 redundantly (discarded at write-back). This keeps the
// panel body a single basic block: 16 back-to-back WMMAs, no exec-mask or
// phi-copy traffic on the 32 accumulator VGPRs.
__device__ __forceinline__ void gemm_panel(const float* hbuf, int kbase,
                                           const float* wpan,
                                           v8f acc[MAX_TILES],
                                           int l15, int kb, int swave) {
  int cofs[MAX_TILES];
#pragma unroll
  for (int j = 0; j < MAX_TILES; ++j) {
    int nt = swave + NWAVES * j;
    nt = (nt < NT_HID) ? nt : (NT_HID - 1);   // branchless clamp (s_min)
    cofs[j] = nt * 16 + l15;
  }
#pragma unroll
  for (int k0 = 0; k0 < PANEL_K; k0 += 4) {
    v2f a = *(const v2f*)(hbuf + l15 * LDH + kbase + k0 + kb);
#pragma unroll
    for (int j = 0; j < MAX_TILES; ++j) {
      v2f b;
      b.x = wpan[(k0 + kb) * HID_DIM + cofs[j]];
      b.y = wpan[(k0 + kb + 1) * HID_DIM + cofs[j]];
      acc[j] = wmma_f32(a, b, acc[j]);
    }
  }
}

// hidden layer: nxt = ELU(LN(cur @ W + b) * g + be), W double-buffered via
// async global->LDS panel copies. Panel loop unrolled by 2 so each stage has
// fixed LDS buffer identities (no parity-dependent pointer selects).
__device__ __forceinline__ void layer_hidden(const float* cur, float* nxt,
                                             float* wp0, float* wp1,
                                             const float* __restrict__ W,
                                             const float* __restrict__ bias,
                                             const float* __restrict__ g,
                                             const float* __restrict__ be,
                                             int din) {
  const int tid   = threadIdx.x;
  const int lane  = tid & 31;
  const int l15   = lane & 15;
  const int kb    = (lane >> 4) << 1;
  const int swave = __builtin_amdgcn_readfirstlane(tid >> 5);  // uniform SGPR

  v8f acc[MAX_TILES];
#pragma unroll
  for (int j = 0; j < MAX_TILES; ++j)
    acc[j] = (v8f){0.f, 0.f, 0.f, 0.f, 0.f, 0.f, 0.f, 0.f};

  const int np = din >> 4;   // panels of K=16 (16 or 25, uniform)
  copy_panel(W, wp0);        // prefetch panel 0

#define PANEL_STAGE(BUFC, BUFN)                                               \
  {                                                                           \
    if (kp + 1 < np) {                                                        \
      copy_panel(W + (size_t)(kp + 1) * PANEL_FLOATS, BUFN);                  \
      WAIT_ASYNC(VECS_PER_THR); /* newest 7 in flight => panel kp arrived */  \
    } else {                                                                  \
      WAIT_ASYNC(0);                                                          \
    }                                                                         \
    __syncthreads(); /* all waves' panel-kp writes visible */                 \
    gemm_panel(cur, kp * PANEL_K, BUFC, acc, l15, kb, swave);                 \
    __syncthreads(); /* done reading: BUFN's twin reusable next stage */      \
    ++kp;                                                                     \
    if (kp >= np) break;                                                      \
  }

  for (int kp = 0;;) {
    PANEL_STAGE(wp0, wp1)
    PANEL_STAGE(wp1, wp0)
  }
#undef PANEL_STAGE

  // write gemm + bias into nxt (pre-LN); scalar guard drops redundant tiles
#pragma unroll
  for (int j = 0; j < MAX_TILES; ++j) {
    if (swave + NWAVES * j < NT_HID) {
      const int c  = (swave + NWAVES * j) * 16 + l15;
      const int rb = (lane >> 4) << 3;   // C/D layout: rows 0..7 / 8..15
      const float bc = bias[c];
#pragma unroll
      for (int v = 0; v < 8; ++v)
        nxt[(rb + v) * LDH + c] = acc[j][v] + bc;
    }
  }
  __syncthreads();

  // LayerNorm + ELU, one wave per row (wave32 shuffle reduction)
  for (int r = swave; r < TM; r += NWAVES) {
    float s = 0.f, s2 = 0.f;
    for (int cc = lane; cc < HID_DIM; cc += 32) {
      float x = nxt[r * LDH + cc];
      s += x; s2 += x * x;
    }
#pragma unroll
    for (int off = 16; off > 0; off >>= 1) {
      s  += __shfl_xor(s,  off, 32);
      s2 += __shfl_xor(s2, off, 32);
    }
    const float mu   = s * (1.f / (float)HID_DIM);
    const float var  = s2 * (1.f / (float)HID_DIM) - mu * mu;
    const float rstd = rsqrtf(var + LN_EPS_C);
    for (int cc = lane; cc < HID_DIM; cc += 32) {
      float x = (nxt[r * LDH + cc] - mu) * rstd * g[cc] + be[cc];
      nxt[r * LDH + cc] = (x > 0.f) ? x : expm1f(x);
    }
  }
  __syncthreads();
}

// final linear tile (dout = 32 or 1): B direct from global (tiny traffic)
__device__ __forceinline__ v8f gemm_tile_global(const float* cur,
                                                const float* __restrict__ W,
                                                int din, int dout, int nt,
                                                int lane) {
  const int  l15 = lane & 15;
  const int  kb  = (lane >> 4) << 1;
  const int  c   = nt * 16 + l15;
  const bool cok = (c < dout);
  v8f acc = {0.f, 0.f, 0.f, 0.f, 0.f, 0.f, 0.f, 0.f};
  for (int k0 = 0; k0 < din; k0 += 4) {
    v2f a = *(const v2f*)(cur + l15 * LDH + k0 + kb);
    v2f b;
    b.x = cok ? W[(size_t)(k0 + kb) * dout + c]     : 0.f;
    b.y = cok ? W[(size_t)(k0 + kb + 1) * dout + c] : 0.f;
    acc = wmma_f32(a, b, acc);
  }
  return acc;
}

// Fully-fused 4xLN-MLP + final linear for a 16-row tile.
__global__ __launch_bounds__(MLP_THREADS)
void mlp_kernel(MlpParams p, const float* __restrict__ X, float* __restrict__ out,
                int nrows, int dout) {
  __shared__ float h0[TM * LDH];
  __shared__ float h1[TM * LDH];
  __shared__ float wp0[PANEL_FLOATS];
  __shared__ float wp1[PANEL_FLOATS];

  const int row0 = blockIdx.x * TM;
  const int tid  = threadIdx.x;

  // stage input rows (F_DIM = 256) into LDS, zero-pad tail rows
  for (int i = tid; i < TM * F_DIM; i += MLP_THREADS) {
    const int r = i >> 8, cc = i & (F_DIM - 1);
    const int gr = row0 + r;
    h0[r * LDH + cc] = (gr < nrows) ? X[(size_t)gr * F_DIM + cc] : 0.f;
  }
  __syncthreads();

  float* cur = h0;
  float* nxt = h1;
  int din = F_DIM;
#pragma unroll
  for (int l = 0; l < 4; ++l) {
    layer_hidden(cur, nxt, wp0, wp1, p.W[l], p.b[l], p.g[l], p.be[l], din);
    float* t = cur; cur = nxt; nxt = t;
    din = HID_DIM;
  }

  // final linear (dout = 32 or 1) straight to global
  const int lane  = tid & 31;
  const int swave = __builtin_amdgcn_readfirstlane(tid >> 5);
  const int NT = (dout + 15) >> 4;
  for (int nt = swave; nt < NT; nt += NWAVES) {   // scalar loop
    v8f acc = gemm_tile_global(cur, p.W[4], HID_DIM, dout, nt, lane);
    const int c  = nt * 16 + (lane & 15);
    const int rb = (lane >> 4) << 3;
    if (c < dout) {                               // per-lane store guard: fine
      const float bc = p.b[4][c];
#pragma unroll
      for (int v = 0; v < 8; ++v) {
        const int gr = row0 + rb + v;
        if (gr < nrows) out[(size_t)gr * dout + c] = acc[v] + bc;
      }
    }
  }
}

// per-row log_softmax over A=32; emits action_logprob and entropy term
__global__ void logsoftmax_kernel(const float* __restrict__ logits,
                                  const float* __restrict__ actions,
                                  float* __restrict__ alp,
                                  float* __restrict__ ent, int nrows) {
  const int i = blockIdx.x * blockDim.x + threadIdx.x;
  if (i >= nrows) return;
  const float* lg = logits + (size_t)i * A_DIM;
  float mx = lg[0];
#pragma unroll
  for (int a = 1; a < A_DIM; ++a) mx = fmaxf(mx, lg[a]);
  float s = 0.f;
#pragma unroll
  for (int a = 0; a < A_DIM; ++a) s += expf(lg[a] - mx);
  const float lse = logf(s);
  float av = 0.f, ev = 0.f;
#pragma unroll
  for (int a = 0; a < A_DIM; ++a) {
    const float lp = lg[a] - mx - lse;
    ev -= expf(lp) * lp;
    av += actions[(size_t)i * A_DIM + a] * lp;
  }
  alp[i] = av;
  ent[i] = ev;
}

// reverse GAE scan + forward reality-weight cumprod + loss reductions.
// One block, one thread per column m (M_DIM = 512).
__global__ __launch_bounds__(M_DIM)
void scan_loss_kernel(const float* __restrict__ rewards,
                      const float* __restrict__ terminals,
                      const float* __restrict__ vc,
                      const float* __restrict__ vt,
                      const float* __restrict__ alp,
                      const float* __restrict__ ent,
                      float* __restrict__ adv_gae,
                      float* __restrict__ out) {
  const int m = threadIdx.x;
  const int T = J_DIM - 1;   // 127

  float agae = 0.f;
  for (int t = T - 1; t >= 0; --t) {
    const float term1 = terminals[(t + 1) * M_DIM + m];
    const float r1    = rewards[(t + 1) * M_DIM + m];
    const float v0t   = vt[t * M_DIM + m];
    const float v1t   = vt[(t + 1) * M_DIM + m];
    const float a = -v0t + r1 + GAMMA_C * (1.f - term1) * v1t;
    agae = a + LAMBDA_C * GAMMA_C * (1.f - term1) * agae;
    adv_gae[t * M_DIM + m] = agae;
  }

  float wgt = 1.f, sv = 0.f, sp = 0.f, se = 0.f;
  for (int t = 0; t < T; ++t) {
    wgt *= (1.f - terminals[t * M_DIM + m]);   // == exp(cumsum(log(1-term0)))
    const float ag = adv_gae[t * M_DIM + m];
    const float vtarget = ag + vt[t * M_DIM + m];
    const float d = vtarget - vc[t * M_DIM + m];
    sv += 0.5f * d * d * wgt;
    sp += -alp[t * M_DIM + m] * ag * wgt;
    se += ent[t * M_DIM + m] * wgt;
  }

  __shared__ float red[M_DIM];
  auto blockReduce = [&](float v) -> float {
    red[m] = v; __syncthreads();
    for (int s = M_DIM / 2; s > 0; s >>= 1) {
      if (m < s) red[m] += red[m + s];
      __syncthreads();
    }
    const float r = red[0];
    __syncthreads();
    return r;
  };
  const float SV = blockReduce(sv);
  const float SP = blockReduce(sp);
  const float SE = blockReduce(se);
  if (m == 0) {
    const float inv = 1.f / (float)(T * M_DIM);
    out[0] = SP * inv - TEMP_C * (SE * inv);  // loss_actor
    out[1] = SV * inv;                        // loss_critic
  }
}

static void fill_params(MlpParams& p, void* const* d_in, int base) {
  for (int l = 0; l < 4; ++l) {
    p.W[l]  = (const float*)d_in[base + 4 * l + 0];
    p.b[l]  = (const float*)d_in[base + 4 * l + 1];
    p.g[l]  = (const float*)d_in[base + 4 * l + 2];
    p.be[l] = (const float*)d_in[base + 4 * l + 3];
  }
  p.W[4] = (const float*)d_in[base + 16];
  p.b[4] = (const float*)d_in[base + 17];
}

extern "C" void kernel_launch(void* const* d_in, const int* in_sizes, int n_in,
                              void* d_out, int out_size, void* d_ws, size_t ws_size,
                              hipStream_t stream) {
  (void)in_sizes; (void)n_in; (void)out_size; (void)ws_size;

  const float* features  = (const float*)d_in[0];  // (J, M, F)
  const float* rewards   = (const float*)d_in[1];  // (J, M)
  const float* terminals = (const float*)d_in[2];  // (J, M)
  const float* actions   = (const float*)d_in[3];  // (J-1, M, A)

  MlpParams pa, pc, pt;
  fill_params(pa, d_in, 4);    // actor   : 18 leaves
  fill_params(pc, d_in, 22);   // critic
  fill_params(pt, d_in, 40);   // target

  const int NA = (J_DIM - 1) * M_DIM;   // 65024 actor rows
  const int NV = J_DIM * M_DIM;         // 65536 value rows

  float* w = (float*)d_ws;
  size_t off = 0;
  float* logits = w + off; off += (size_t)NA * A_DIM;
  float* vc     = w + off; off += (size_t)NV;
  float* vt     = w + off; off += (size_t)NV;
  float* alp    = w + off; off += (size_t)NA;
  float* ent    = w + off; off += (size_t)NA;
  float* adv    = w + off;               // (J-1, M)

  mlp_kernel<<<(NA + TM - 1) / TM, MLP_THREADS, 0, stream>>>(pa, features, logits, NA, A_DIM);
  mlp_kernel<<<(NV + TM - 1) / TM, MLP_THREADS, 0, stream>>>(pc, features, vc, NV, 1);
  mlp_kernel<<<(NV + TM - 1) / TM, MLP_THREADS, 0, stream>>>(pt, features, vt, NV, 1);

  logsoftmax_kernel<<<(NA + 255) / 256, 256, 0, stream>>>(logits, actions, alp, ent, NA);

  scan_loss_kernel<<<1, M_DIM, 0, stream>>>(rewards, terminals, vc, vt, alp, ent,
                                            adv, (float*)d_out);
}